// LabelLoss_39058432590060
// MI455X (gfx1250) — compile-verified
//
#include <hip/hip_runtime.h>

// Problem constants (from reference setup_inputs)
#define B_   128
#define C_   16
#define H_   512
#define W_   512
#define HW_  (H_ * W_)
#define SPLIT 16                 // tiles per sample for the argmax pass
#define CHUNK (HW_ / SPLIT)      // 16384 floats per tile
#define T1   256                 // threads per argmax block (8 wave32s)

typedef __attribute__((ext_vector_type(2))) float v2f;
typedef __attribute__((ext_vector_type(8))) float v8f;

// ---------------------------------------------------------------------------
// K1: streaming partial argmax. grid = B_*SPLIT blocks; each block scans
// 64 KB of one sample with coalesced float4 loads. First-occurrence tiebreak.
// ---------------------------------------------------------------------------
__global__ __launch_bounds__(T1)
void argmax_partial(const float* __restrict__ heat,
                    float* __restrict__ pVal,
                    int*   __restrict__ pIdx) {
  const int blk = blockIdx.x;
  const int b   = blk / SPLIT;
  const int s   = blk % SPLIT;
  const float* base = heat + (size_t)b * HW_ + (size_t)s * CHUNK;
  const int tid = threadIdx.x;

  float bestV = -__builtin_huge_valf();
  int   bestI = 0x7FFFFFFF;

  // Each thread: 16 float4 loads, flat indices strictly ascending per thread,
  // so strict '>' preserves first-occurrence semantics within the thread.
  #pragma unroll
  for (int i = 0; i < CHUNK / (T1 * 4); ++i) {
    const int e  = (i * T1 + tid) * 4;
    const float4 v = *(const float4*)(base + e);
    const int fi = s * CHUNK + e;
    if (v.x > bestV) { bestV = v.x; bestI = fi;     }
    if (v.y > bestV) { bestV = v.y; bestI = fi + 1; }
    if (v.z > bestV) { bestV = v.z; bestI = fi + 2; }
    if (v.w > bestV) { bestV = v.w; bestI = fi + 3; }
  }

  __shared__ float sv[T1];
  __shared__ int   si[T1];
  sv[tid] = bestV;
  si[tid] = bestI;
  __syncthreads();

  // Tree reduction; on equal value keep the smaller flat index (first occurrence).
  for (int off = T1 / 2; off > 0; off >>= 1) {
    if (tid < off) {
      const float ov = sv[tid + off];
      const int   oi = si[tid + off];
      if (ov > sv[tid] || (ov == sv[tid] && oi < si[tid])) {
        sv[tid] = ov; si[tid] = oi;
      }
    }
    __syncthreads();
  }
  if (tid == 0) { pVal[blk] = sv[0]; pIdx[blk] = si[0]; }
}

// ---------------------------------------------------------------------------
// K2: finalize. One block, 128 threads (thread t == sample b).
//  Phase A: fold SPLIT partials -> (x,y) per sample.
//  Phase B: gather pred, squared diffs -> LDS (128x16 = 2048 floats).
//  Phase C: wave 0 only: 32x V_WMMA_F32_16X16X4_F32 with all-ones B matrix.
//           Since B==1, every D[i][j] is a row-sum, so sum(all D) = 16*total.
// ---------------------------------------------------------------------------
__global__ __launch_bounds__(128)
void finalize_wmma(const float* __restrict__ pred,
                   const float* __restrict__ gt,
                   const float* __restrict__ pVal,
                   const int*   __restrict__ pIdx,
                   float* __restrict__ out) {
  __shared__ float sq[B_ * C_];   // 8 KB of squared diffs

  const int t = threadIdx.x;      // 0..127 == sample id

  // Phase A: per-sample argmax over SPLIT partials (first-occurrence tiebreak)
  float bv = -__builtin_huge_valf();
  int   bi = 0x7FFFFFFF;
  #pragma unroll
  for (int s = 0; s < SPLIT; ++s) {
    const float v = pVal[t * SPLIT + s];
    const int   i = pIdx[t * SPLIT + s];
    if (v > bv || (v == bv && i < bi)) { bv = v; bi = i; }
  }
  const int x = bi / W_;          // row  (flat_idx // W)
  const int y = bi % W_;          // col  (flat_idx %  W)

  // Phase B: squared differences for all 16 channels of this sample
  const float* pxy = pred + (size_t)x * W_ + y;
  #pragma unroll
  for (int c = 0; c < C_; ++c) {
    const float d = pxy[(size_t)c * HW_] - gt[t * C_ + c];
    sq[t * C_ + c] = d * d;
  }
  __syncthreads();

  // Phase C: wave 0 (lanes 0..31, EXEC all ones) reduces via WMMA
  if (t < 32) {
    // A-matrix 16x4 f32 layout: lane L holds row M = L&15; VGPR pair holds
    // K = {0,1} for lanes 0-15 and K = {2,3} for lanes 16-31.
    const int M  = t & 15;
    const int k0 = (t >> 4) * 2;

    v8f acc = {};
    v2f bones; bones[0] = 1.0f; bones[1] = 1.0f;   // 4x16 all-ones B

    #pragma unroll
    for (int ch = 0; ch < (B_ * C_) / 64; ++ch) {  // 32 chunks of 64 values
      const float* ap = &sq[ch * 64 + M * 4 + k0];
      v2f a; a[0] = ap[0]; a[1] = ap[1];
      // 8 args: (neg_a, A, neg_b, B, c_mod, C, reuse_a, reuse_b)
      acc = __builtin_amdgcn_wmma_f32_16x16x4_f32(
          /*neg_a=*/false, a, /*neg_b=*/false, bones,
          /*c_mod=*/(short)0, acc, /*reuse_a=*/false, /*reuse_b=*/false);
    }

    // Sum of all 256 accumulator entries == 16 * grand total.
    float part = acc[0] + acc[1] + acc[2] + acc[3] +
                 acc[4] + acc[5] + acc[6] + acc[7];
    #pragma unroll
    for (int m = 16; m > 0; m >>= 1) part += __shfl_xor(part, m, 32);

    if (t == 0) out[0] = part * (1.0f / 16.0f);
  }
}

// ---------------------------------------------------------------------------
extern "C" void kernel_launch(void* const* d_in, const int* in_sizes, int n_in,
                              void* d_out, int out_size, void* d_ws, size_t ws_size,
                              hipStream_t stream) {
  const float* pred = (const float*)d_in[0];  // [16, 512, 512]
  const float* gt   = (const float*)d_in[1];  // [128, 16]
  const float* heat = (const float*)d_in[2];  // [128, 512, 512]

  float* pVal = (float*)d_ws;                                   // B_*SPLIT floats
  int*   pIdx = (int*)((char*)d_ws + B_ * SPLIT * sizeof(float)); // B_*SPLIT ints

  argmax_partial<<<B_ * SPLIT, T1, 0, stream>>>(heat, pVal, pIdx);
  finalize_wmma<<<1, 128, 0, stream>>>(pred, gt, pVal, pIdx, (float*)d_out);
}